// PPYoloELoss_45397804319344
// MI455X (gfx1250) — compile-verified
//
#include <hip/hip_runtime.h>
#include <hip/hip_bf16.h>

// ---------------------------------------------------------------------------
// PPYOLOE / ATSS task-aligned assigner for MI455X (gfx1250, wave32).
//
// K1: pairwise squared center distance as a rank-4 GEMM on the matrix core:
//       d^2 = [ax, ay, 1, a^2] . [-2gx, -2gy, g^2, 1]^T
//     -> V_WMMA_F32_16X16X4_F32 (16 anchors x 16 gts per op).
//     Top-9 selection is fully branchless: (d2,idx) packed into a u64 key,
//     sorted-9 maintained with v_cmp_lt_u64 + cndmask min/max chains
//     (round-1's if/swap version caused EXEC-branch mov storms).
//     Levels are split into 11 chunks of <=800 anchors for WGP parallelism;
//     each lane dumps its partial sorted-9 keys to workspace.
// K1b: merge <=144 partial keys -> 9 candidates per (gt, level).
// K2:  27 candidate IoUs -> thr = mean+std; scatter candidate bits.
// K3:  streaming assign + one-hot score write (the 46 MB memory floor pass).
// ---------------------------------------------------------------------------

typedef __attribute__((ext_vector_type(2))) float v2f;
typedef __attribute__((ext_vector_type(8))) float v8f;
typedef unsigned long long u64;

#define TOPK 9
#define NLEVELS 3
#define NCHUNK 11         // 8 (level0) + 2 (level1) + 1 (level2)
#define PERCHUNK (2 * TOPK)   // two half-column lists (lanes n and n+16)
#define EPS_IOU 1e-9f
#define MAX_M 128

__device__ __forceinline__ float iou_box(float ax1, float ay1, float ax2, float ay2,
                                         float bx1, float by1, float bx2, float by2) {
    float area1 = (ax2 - ax1) * (ay2 - ay1);
    float area2 = (bx2 - bx1) * (by2 - by1);
    float ix1 = fmaxf(ax1, bx1);
    float iy1 = fmaxf(ay1, by1);
    float ix2 = fminf(ax2, bx2);
    float iy2 = fminf(ay2, by2);
    float inter = fmaxf(ix2 - ix1, 0.0f) * fmaxf(iy2 - iy1, 0.0f);
    float uni = area1 + area2 - inter;
    return inter / (uni + EPS_IOU);
}

__device__ __forceinline__ float iou_f4(const float4 a, const float4 g) {
    return iou_box(a.x, a.y, a.z, a.w, g.x, g.y, g.z, g.w);
}

// Branchless sorted-ascending insert of `key` into k9[TOPK].
__device__ __forceinline__ void insert9(u64* k9, u64 key) {
#pragma unroll
    for (int j = 0; j < TOPK; ++j) {
        bool lt = key < k9[j];
        u64 mn = lt ? key : k9[j];
        u64 mx = lt ? k9[j] : key;
        k9[j] = mn;
        key = mx;
    }
}

// ---------------------------------------------------------------------------
// Kernel 1: grid (NCHUNK, B), 256 threads (8 waves); wave w owns gt tile w.
// Streams this chunk's anchor tiles through WMMA; each lane keeps a sorted
// top-9 of u64 (d2bits, anchor_idx) keys for its half-column of the D tile,
// then writes the 9 keys to pkeys.  No LDS, no divergence around the WMMA.
// ---------------------------------------------------------------------------
__global__ __launch_bounds__(256) void topk_partial(
    const float4* __restrict__ anchors,   // [N]
    const float4* __restrict__ gts,       // [B*M]
    u64* __restrict__ pkeys,              // [B*M][NCHUNK][PERCHUNK]
    int M)
{
    const int c    = blockIdx.x;          // chunk 0..10
    const int b    = blockIdx.y;
    const int lane = threadIdx.x & 31;
    const int wave = threadIdx.x >> 5;
    const int col  = lane & 15;
    const bool hi  = lane >= 16;

    const int astart = (c < 8) ? (c * 800) : ((c < 10) ? (6400 + (c - 8) * 800) : 8000);
    const int ntiles = (c < 10) ? 50 : 25;

    const int  m       = wave * 16 + col;
    const bool valid_m = (m < M);
    const int  mc      = valid_m ? m : (M - 1);

    // B operand: column = gt center trick vector [-2gx, -2gy, g2, 1]
    float4 gb  = gts[(size_t)b * M + mc];
    float  gcx = 0.5f * (gb.x + gb.z);
    float  gcy = 0.5f * (gb.y + gb.w);
    float  g2  = gcx * gcx + gcy * gcy;
    v2f bop;
    if (!hi) { bop.x = -2.0f * gcx; bop.y = -2.0f * gcy; }  // K=0,1
    else     { bop.x = g2;          bop.y = 1.0f;        }  // K=2,3

    u64 k9[TOPK];
#pragma unroll
    for (int j = 0; j < TOPK; ++j) k9[j] = ~0ull;

    v8f czero = {};

    for (int t = 0; t < ntiles; ++t) {
        // A operand: row = anchor trick vector [ax, ay, 1, a2]
        int    abase = astart + t * 16;
        float4 ab    = anchors[abase + col];
        float  acx   = 0.5f * (ab.x + ab.z);
        float  acy   = 0.5f * (ab.y + ab.w);
        float  a2    = acx * acx + acy * acy;
        v2f aop;
        if (!hi) { aop.x = acx;  aop.y = acy; }   // K=0,1
        else     { aop.x = 1.0f; aop.y = a2;  }   // K=2,3

        v8f d = __builtin_amdgcn_wmma_f32_16x16x4_f32(
            false, aop, false, bop, (short)0, czero, false, false);

        // This lane holds gt column (lane&15), anchor rows r(+8 if hi).
#pragma unroll
        for (int r = 0; r < 8; ++r) {
            float d2 = fmaxf(d[r], 0.0f);   // clamp: bitcast stays monotonic
            unsigned int gidx = (unsigned int)(abase + (hi ? (r + 8) : r));
            u64 key = ((u64)__float_as_uint(d2) << 32) | gidx;
            insert9(k9, key);
        }
    }

    if (valid_m) {
        u64* dst = pkeys + (((size_t)b * M + m) * NCHUNK + c) * PERCHUNK + (hi ? TOPK : 0);
#pragma unroll
        for (int j = 0; j < TOPK; ++j) dst[j] = k9[j];
    }
}

// ---------------------------------------------------------------------------
// Kernel 1b: one thread per (b, m, level); select the 9 smallest keys of this
// level's chunks (<=144 keys) -> candidate anchor indices.
// ---------------------------------------------------------------------------
__global__ void topk_merge(
    const u64* __restrict__ pkeys,        // [B*M][NCHUNK][PERCHUNK]
    int* __restrict__ cand,               // [B*M][27]
    int BM)
{
    int idx = blockIdx.x * blockDim.x + threadIdx.x;
    if (idx >= BM * NLEVELS) return;
    int bm  = idx / NLEVELS;
    int lvl = idx - bm * NLEVELS;

    int c0 = (lvl == 0) ? 0 : ((lvl == 1) ? 8 : 10);
    int nc = (lvl == 0) ? 8 : ((lvl == 1) ? 2 : 1);

    u64 k9[TOPK];
#pragma unroll
    for (int j = 0; j < TOPK; ++j) k9[j] = ~0ull;

    const u64* src = pkeys + ((size_t)bm * NCHUNK + c0) * PERCHUNK;
    for (int e = 0; e < nc * PERCHUNK; ++e) insert9(k9, src[e]);

    int* dst = cand + (size_t)bm * (NLEVELS * TOPK) + lvl * TOPK;
#pragma unroll
    for (int j = 0; j < TOPK; ++j) dst[j] = (int)(unsigned int)(k9[j] & 0xffffffffull);
}

// ---------------------------------------------------------------------------
// Kernel 2: one thread per (b, m). 27 on-the-fly candidate IoUs ->
// thr = mean + std (count is always 27: distinct anchors, disjoint levels).
// Also scatter candidate bits into the per-(b,n) 128-bit gt mask.
// ---------------------------------------------------------------------------
__global__ void thresh_kernel(
    const float4* __restrict__ anchors,
    const float4* __restrict__ gts,
    const int* __restrict__ cand,
    float* __restrict__ thr,
    unsigned int* __restrict__ candbits,   // [B*N*4] words, pre-zeroed
    int B, int M, int N)
{
    int idx = blockIdx.x * blockDim.x + threadIdx.x;
    if (idx >= B * M) return;
    int b = idx / M;
    int m = idx - b * M;

    float4 g = gts[idx];
    const int* cc = cand + (size_t)idx * (NLEVELS * TOPK);
    unsigned int bit  = 1u << (m & 31);
    int          word = m >> 5;

    float sum = 0.0f, sumsq = 0.0f;
#pragma unroll
    for (int j = 0; j < NLEVELS * TOPK; ++j) {
        int n = cc[j];
        float v = iou_f4(anchors[n], g);
        sum += v;
        sumsq += v * v;
        atomicOr(&candbits[((size_t)b * N + n) * 4 + word], bit);
    }
    const float inv = 1.0f / (float)(NLEVELS * TOPK);
    float mean = sum * inv;
    float var  = sumsq * inv - mean * mean;
    thr[idx] = mean + sqrtf(fmaxf(var, 0.0f));
}

// ---------------------------------------------------------------------------
// Kernel 3: one thread per (b, n). GT data staged in LDS; iterate set bits of
// the candidate mask, apply threshold/center/mask_gt tests, argmax over gts,
// then emit labels / boxes / one-hot scores / fg in one streaming pass.
// ---------------------------------------------------------------------------
__global__ __launch_bounds__(256) void assign_kernel(
    const float4* __restrict__ anchors,
    const int* __restrict__ labels,
    const float4* __restrict__ gts,
    const int* __restrict__ maskgt,
    const float4* __restrict__ preds,
    const unsigned int* __restrict__ candbits,
    const float* __restrict__ thr,
    float* __restrict__ out,
    int B, int M, int N, int C)
{
    const int b = blockIdx.y;
    const int n = blockIdx.x * 256 + threadIdx.x;

    __shared__ float4 sg[MAX_M];
    __shared__ float  sthr[MAX_M];
    __shared__ int    slab[MAX_M];
    __shared__ int    smsk[MAX_M];
    for (int i = threadIdx.x; i < M && i < MAX_M; i += 256) {
        sg[i]   = gts[(size_t)b * M + i];
        sthr[i] = thr[b * M + i];
        slab[i] = labels[b * M + i];
        smsk[i] = maskgt[b * M + i];
    }
    __syncthreads();
    if (n >= N) return;

    float4 ab  = anchors[n];
    float  acx = 0.5f * (ab.x + ab.z);
    float  acy = 0.5f * (ab.y + ab.w);

    const unsigned int* cb = candbits + ((size_t)b * N + n) * 4;
    float best = 0.0f;
    int   bm   = 0;
    bool  fg   = false;

    const int nw = (M + 31) >> 5;
    for (int w = 0; w < nw; ++w) {
        unsigned int wordv = cb[w];
        while (wordv) {
            int t = __ffs(wordv) - 1;
            wordv &= wordv - 1;
            int m = (w << 5) + t;
            float4 g = sg[m];
            float  v = iou_f4(ab, g);
            bool cin = (acx >= g.x) && (acx <= g.z) && (acy >= g.y) && (acy <= g.w);
            if ((v >= sthr[m]) && cin && (smsk[m] > 0)) {
                fg = true;
                if (v > best) { best = v; bm = m; }
            }
        }
    }

    const size_t bn = (size_t)b * N + n;
    const size_t o1 = (size_t)B * N;
    float* out_lab = out;                             // [B*N]
    float* out_box = out + o1;                        // [B*N*4]
    float* out_sc  = out + o1 * 5;                    // [B*N*C]
    float* out_fg  = out + o1 * 5 + o1 * (size_t)C;   // [B*N]

    float4 g    = sg[bm];
    int    lab  = fg ? slab[bm] : 0;
    float  aiou = fg ? iou_f4(ab, g) : 0.0f;
    float  piou = fg ? iou_f4(preds[bn], g) : 0.0f;
    float  s    = aiou * piou;   // fg-gated already

    out_lab[bn] = (float)lab;
    float4 obox = fg ? g : make_float4(0.0f, 0.0f, 0.0f, 0.0f);
    ((float4*)out_box)[bn] = obox;
    out_fg[bn] = fg ? 1.0f : 0.0f;

    float* sc = out_sc + bn * (size_t)C;
    if ((C & 3) == 0) {
        for (int j4 = 0; j4 < C; j4 += 4) {
            float4 v = make_float4(0.0f, 0.0f, 0.0f, 0.0f);
            if (lab >= j4 && lab < j4 + 4) (&v.x)[lab - j4] = s;
            *(float4*)(sc + j4) = v;
        }
    } else {
        for (int j = 0; j < C; ++j) sc[j] = (j == lab) ? s : 0.0f;
    }
}

// ---------------------------------------------------------------------------
// Host launcher. Scratch (d_ws) layout (8-byte aligned first):
//   pkeys    : B*M*NCHUNK*PERCHUNK u64  (~3.25 MB)
//   candbits : B*N*4 u32               (~2.15 MB, memset to 0)
//   cand     : B*M*27 int              (~216 KB)
//   thr      : B*M float               (~8 KB)
// ---------------------------------------------------------------------------
extern "C" void kernel_launch(void* const* d_in, const int* in_sizes, int n_in,
                              void* d_out, int out_size, void* d_ws, size_t ws_size,
                              hipStream_t stream) {
    const float4* anchors = (const float4*)d_in[0];     // [N,4] f32
    const int*    gtlab   = (const int*)d_in[1];        // [B,M,1] int
    const float4* gtbox   = (const float4*)d_in[2];     // [B,M,4] f32
    const int*    maskgt  = (const int*)d_in[3];        // [B,M,1] int
    const float4* preds   = (const float4*)d_in[4];     // [B,N,4] f32

    const int N  = in_sizes[0] / 4;
    const int B  = in_sizes[4] / (4 * N);
    const int M  = in_sizes[1] / B;
    const int C  = out_size / (B * N) - 6;

    u64*   pkeys      = (u64*)d_ws;
    size_t pkeysBytes = (size_t)B * M * NCHUNK * PERCHUNK * sizeof(u64);
    unsigned int* candbits = (unsigned int*)((char*)d_ws + pkeysBytes);
    size_t cbBytes    = (size_t)B * N * 4 * sizeof(unsigned int);
    int*   cand       = (int*)((char*)candbits + cbBytes);
    size_t candBytes  = (size_t)B * M * (NLEVELS * TOPK) * sizeof(int);
    float* thr        = (float*)((char*)cand + candBytes);

    hipMemsetAsync(candbits, 0, cbBytes, stream);

    dim3 g1(NCHUNK, B);
    topk_partial<<<g1, 256, 0, stream>>>(anchors, gtbox, pkeys, M);

    int nmerge = B * M * NLEVELS;
    topk_merge<<<(nmerge + 127) / 128, 128, 0, stream>>>(pkeys, cand, B * M);

    int nbm = B * M;
    thresh_kernel<<<(nbm + 127) / 128, 128, 0, stream>>>(
        anchors, gtbox, cand, thr, candbits, B, M, N);

    dim3 g3((N + 255) / 256, B);
    assign_kernel<<<g3, 256, 0, stream>>>(
        anchors, gtlab, gtbox, maskgt, preds, candbits, thr,
        (float*)d_out, B, M, N, C);
}